// SideWindowFilter_48919677501779
// MI455X (gfx1250) — compile-verified
//
#include <hip/hip_runtime.h>
#include <hip/hip_bf16.h>

#define RADIUS 3
#define ITER   6
#define IMH    512
#define IMW    512
#define NIMG   12          // b*c = 4*3, channels folded into batch

#define LDW     24         // padded halo/tile stride; 24 floats => conflict-free split-half loads
#define S_ELEMS (LDW*LDW)  // 576
#define V_ELEMS (16*LDW)   // 384
#define WAVE_F  (S_ELEMS + 2*V_ELEMS)  // floats of LDS per wave

typedef __attribute__((ext_vector_type(2))) float v2f;
typedef __attribute__((ext_vector_type(8))) float v8f;

__device__ __forceinline__ v8f wmma4(v2f a, v2f b, v8f c) {
    // D = A(16x4,f32) * B(4x16,f32) + C(16x16,f32)  -- exact f32 path
    return __builtin_amdgcn_wmma_f32_16x16x4_f32(
        false, a, false, b, (short)0, c, false, false);
}

struct DirSums {
    float nw[8], ne[8], swq[8], seq[8];   // quarter-window sums (C layout, 8 elems/lane)
    float hls[8], hrs[8];                 // horiz L/R sums of center rows
    float vuc[8], vdc[8];                 // vertical up/down sums at center col
    float z[8];                           // center pixel
};

// Banded 0/1 weights, shared by vertical-A and horizontal-B operands:
// WL[t] = band [m, m+3], WR[t] = band [m+3, m+6] at K = 4t + 2*khalf (+0/+1).
struct BandW {
    v2f wl[6];
    v2f wr[6];
};

__device__ __forceinline__ void makeBands(int lane, BandW& w) {
    const int m = lane & 15;
    const int khalf = lane >> 4;
    #pragma unroll
    for (int t = 0; t < 6; ++t) {
        int k0 = 4 * t + 2 * khalf;
        w.wl[t].x = (k0     >= m     && k0     <= m + 3) ? 1.f : 0.f;
        w.wl[t].y = (k0 + 1 >= m     && k0 + 1 <= m + 3) ? 1.f : 0.f;
        w.wr[t].x = (k0     >= m + 3 && k0     <= m + 6) ? 1.f : 0.f;
        w.wr[t].y = (k0 + 1 >= m + 3 && k0 + 1 <= m + 6) ? 1.f : 0.f;
        // Opaque pass-through: asm results are NOT rematerializable, so the
        // allocator must keep these 12 VGPR pairs live instead of re-emitting
        // the cmp/cndmask chains at each of the 120 WMMA sites.
        asm volatile("" : "+v"(w.wl[t]));
        asm volatile("" : "+v"(w.wr[t]));
    }
}

// Compute all directional raw sums for one image tile via f32 WMMA.
// S/VU/VD are this wave's private LDS regions.
__device__ void computeDirs(const float* __restrict__ gin, int n, int y0, int x0,
                            float* __restrict__ S, float* __restrict__ VU,
                            float* __restrict__ VD, int lane, const BandW& w,
                            DirSums& o)
{
    const float* img = gin + (size_t)n * IMH * IMW;

    // ---- stage 24x24 halo (zero OOB == reference zero-pad conv) ----
    for (int i = lane; i < S_ELEMS; i += 32) {
        int r = i / LDW, c = i % LDW;
        int y = y0 - RADIUS + r, x = x0 - RADIUS + c;
        float v = 0.f;
        if (y >= 0 && y < IMH && x >= 0 && x < IMW) v = img[y * IMW + x];
        S[i] = v;
    }

    const int m     = lane & 15;   // A row / B col / C col index for this lane
    const int khalf = lane >> 4;   // K-pair selector per 7.12.2 layouts

    // ---- vertical pass: VU = Wup*S, VD = Wdown*S over block cols 0..23 ----
    #pragma unroll
    for (int g = 0; g < 2; ++g) {
        v8f accU = {0,0,0,0,0,0,0,0};
        v8f accD = {0,0,0,0,0,0,0,0};
        const int col = 16 * g + m;
        const bool cok = (col < LDW);
        #pragma unroll
        for (int t = 0; t < 6; ++t) {
            int k0 = 4 * t + 2 * khalf;
            v2f b;
            b.x = cok ? S[k0 * LDW + col]       : 0.f;
            b.y = cok ? S[(k0 + 1) * LDW + col] : 0.f;
            accU = wmma4(w.wl[t], b, accU);
            accD = wmma4(w.wr[t], b, accD);
        }
        // C layout -> LDS (only cols < 24 exist)
        if (16 * g + m < LDW) {
            int cc = 16 * g + m;
            #pragma unroll
            for (int j = 0; j < 8; ++j) {
                int row = j + 8 * khalf;
                VU[row * LDW + cc] = accU[j];
                VD[row * LDW + cc] = accD[j];
            }
        }
    }

    // ---- horizontal pass: HL/HR of VU, VD and S center rows ----
    v8f NW = {0,0,0,0,0,0,0,0};
    v8f NE = NW, SWq = NW, SEq = NW, HLS = NW, HRS = NW;
    #pragma unroll
    for (int t = 0; t < 6; ++t) {
        int k0 = 4 * t + 2 * khalf;
        v2f aU; aU.x = VU[m * LDW + k0]; aU.y = VU[m * LDW + k0 + 1];
        NW  = wmma4(aU, w.wl[t], NW);
        NE  = wmma4(aU, w.wr[t], NE);
        v2f aD; aD.x = VD[m * LDW + k0]; aD.y = VD[m * LDW + k0 + 1];
        SWq = wmma4(aD, w.wl[t], SWq);
        SEq = wmma4(aD, w.wr[t], SEq);
        v2f aS; aS.x = S[(m + RADIUS) * LDW + k0];
        aS.y = S[(m + RADIUS) * LDW + k0 + 1];
        HLS = wmma4(aS, w.wl[t], HLS);
        HRS = wmma4(aS, w.wr[t], HRS);
    }

    // ---- gather per-element centers (C layout: lane holds N=m, M=j+8*khalf) ----
    #pragma unroll
    for (int j = 0; j < 8; ++j) {
        int row = j + 8 * khalf;
        o.vuc[j] = VU[row * LDW + m + RADIUS];
        o.vdc[j] = VD[row * LDW + m + RADIUS];
        o.z[j]   = S[(row + RADIUS) * LDW + m + RADIUS];
        o.nw[j]  = NW[j];  o.ne[j]  = NE[j];
        o.swq[j] = SWq[j]; o.seq[j] = SEq[j];
        o.hls[j] = HLS[j]; o.hrs[j] = HRS[j];
    }
}

__device__ __forceinline__ void makeD(const DirSums& s, int j, float z, float d[8]) {
    const float inv28 = 1.f / 28.f, inv16 = 1.f / 16.f;
    d[0] = (s.nw[j] + s.swq[j] - s.hls[j]) * inv28 - z;  // L
    d[1] = (s.ne[j] + s.seq[j] - s.hrs[j]) * inv28 - z;  // R
    d[2] = (s.nw[j] + s.ne[j]  - s.vuc[j]) * inv28 - z;  // U
    d[3] = (s.swq[j] + s.seq[j] - s.vdc[j]) * inv28 - z; // D
    d[4] = s.nw[j]  * inv16 - z;                         // NW
    d[5] = s.ne[j]  * inv16 - z;                         // NE
    d[6] = s.swq[j] * inv16 - z;                         // SW
    d[7] = s.seq[j] * inv16 - z;                         // SE
}

__global__ __launch_bounds__(256) void SideWindowFilter_iter(
    const float* __restrict__ imIn, const float* __restrict__ peIn,
    float* __restrict__ imOut, float* __restrict__ peOut)
{
    __shared__ float smem[8 * WAVE_F];   // 43008 B, per-wave private regions
    const int wave = threadIdx.x >> 5;
    const int lane = threadIdx.x & 31;
    float* base = smem + wave * WAVE_F;
    float* S  = base;
    float* VU = base + S_ELEMS;
    float* VD = base + S_ELEMS + V_ELEMS;

    const int n  = blockIdx.z;
    const int y0 = blockIdx.y * 16;
    const int x0 = (blockIdx.x * 8 + wave) * 16;

    BandW w;
    makeBands(lane, w);   // banded weights computed ONCE, pinned in VGPRs

    // ---- image pass: d-values, argmin (first-min-wins like jnp.argmin) ----
    DirSums si;
    computeDirs(imIn, n, y0, x0, S, VU, VD, lane, w, si);
    int idx[8]; float dm[8]; float zim[8];
    #pragma unroll
    for (int j = 0; j < 8; ++j) {
        float z = si.z[j];
        float d[8];
        makeD(si, j, z, d);
        int bi = 0; float bv = fabsf(d[0]); float bd = d[0];
        #pragma unroll
        for (int i = 1; i < 8; ++i) {
            float a = fabsf(d[i]);
            if (a < bv) { bv = a; bi = i; bd = d[i]; }
        }
        idx[j] = bi; dm[j] = bd; zim[j] = z;
    }

    // ---- perturbation pass: select dd at im's argmin index ----
    DirSums sp;
    computeDirs(peIn, n, y0, x0, S, VU, VD, lane, w, sp);
    const int khalf = lane >> 4, nn = lane & 15;
    #pragma unroll
    for (int j = 0; j < 8; ++j) {
        float z = sp.z[j];
        float e[8];
        makeD(sp, j, z, e);
        float sel = e[0];
        #pragma unroll
        for (int i = 1; i < 8; ++i) sel = (idx[j] == i) ? e[i] : sel;
        int row = y0 + j + 8 * khalf;
        int col = x0 + nn;
        size_t off = (size_t)n * IMH * IMW + (size_t)row * IMW + col;
        peOut[off] = z + sel;
        imOut[off] = zim[j] + dm[j];
    }
}

extern "C" void kernel_launch(void* const* d_in, const int* in_sizes, int n_in,
                              void* d_out, int out_size, void* d_ws, size_t ws_size,
                              hipStream_t stream) {
    (void)in_sizes; (void)n_in; (void)out_size; (void)ws_size;
    const float* im0 = (const float*)d_in[0];
    const float* pe0 = (const float*)d_in[1];
    float* out = (float*)d_out;
    const size_t SZ = (size_t)NIMG * IMH * IMW;
    float* ws0 = (float*)d_ws;       // im  ping
    float* ws1 = ws0 + SZ;           // pert ping
    float* ws2 = ws1 + SZ;           // im  pong   (pert pong lives in d_out)

    dim3 grid(IMW / 128, IMH / 16, NIMG);  // 8 tiles of 16 cols per block
    dim3 block(256);

    const float* ic = im0;
    const float* pc = pe0;
    for (int it = 0; it < ITER; ++it) {
        float* io; float* po;
        if ((it & 1) == 0) { io = ws0; po = ws1; }
        else               { io = ws2; po = out; }
        SideWindowFilter_iter<<<grid, block, 0, stream>>>(ic, pc, io, po);
        ic = io; pc = po;
    }
    // ITER=6 even -> last write was (ws2, d_out): final pert lands in d_out.
}